// SpikingEyeTracker_33285996544149
// MI455X (gfx1250) — compile-verified
//
#include <hip/hip_runtime.h>

typedef float v8f __attribute__((ext_vector_type(8)));
typedef float v2f __attribute__((ext_vector_type(2)));
typedef int   v4i __attribute__((ext_vector_type(4)));

typedef __attribute__((address_space(1))) v4i* global_v4i_ptr;
typedef __attribute__((address_space(3))) v4i* lds_v4i_ptr;

#define B_  2048
#define C_  5
#define T_  2048
#define H_  16
#define O_  2
#define TS  16    // timesteps staged per LDS chunk
#define TSP 20    // padded row (floats): 80B rows -> 16B aligned, 16 distinct banks

#define NCHUNK (T_ / TS)
#define QUADS_PER_THREAD ((64 * C_ * (TS / 4)) / 128)   // 1280/128 = 10

// ---- async global->LDS copy (CDNA5 GLOBAL_LOAD_ASYNC_TO_LDS_B128, ASYNCcnt) ----
static __device__ __forceinline__ void async_copy_b128(const float* src, float* dst_lds) {
#if __has_builtin(__builtin_amdgcn_global_load_async_to_lds_b128)
    float* s = const_cast<float*>(src);
    __builtin_amdgcn_global_load_async_to_lds_b128(
        (global_v4i_ptr)s, (lds_v4i_ptr)dst_lds, 0, 0);
#else
    unsigned lds_off = (unsigned)(size_t)(lds_v4i_ptr)dst_lds;
    asm volatile("global_load_async_to_lds_b128 %0, %1, off"
                 :: "v"(lds_off), "v"(src) : "memory");
#endif
}

static __device__ __forceinline__ void wait_async0() {
#if __has_builtin(__builtin_amdgcn_s_wait_asynccnt)
    __builtin_amdgcn_s_wait_asynccnt(0);
#else
    asm volatile("s_wait_asynccnt 0" ::: "memory");
#endif
}

// 128 threads = 4 wave32; each wave owns a 16-batch WMMA tile; block owns 64 batches.
__global__ __launch_bounds__(128)
void snn_spike_kernel(const float* __restrict__ x,
                      const float* __restrict__ conv_w,
                      const float* __restrict__ conv_b,
                      const float* __restrict__ bn_gamma,
                      const float* __restrict__ bn_beta,
                      const float* __restrict__ bn_mean,
                      const float* __restrict__ bn_var,
                      const float* __restrict__ fc_w,
                      const float* __restrict__ fc_b,
                      const float* __restrict__ beta1p,
                      const float* __restrict__ beta2p,
                      float* __restrict__ out)
{
    __shared__ __align__(16) float sx[2][64 * C_ * TSP];   // 2 x 25 KB double buffer

    const int tid  = threadIdx.x;
    const int lane = tid & 31;
    const int wave = tid >> 5;
    const int half = lane >> 4;    // 0: lanes 0-15, 1: lanes 16-31
    const int n    = lane & 15;    // col / row-in-half index

    const int bBlock = blockIdx.x * 64;   // 32 blocks cover B=2048
    const int b0     = bBlock + wave * 16;
    const int tb     = wave * 16;         // tile's batch base inside LDS

    // ---- fold conv(middle tap) * 2 and BN into one affine (per lane, h = n) ----
    const float sc   = bn_gamma[n] * rsqrtf(bn_var[n] + 1e-5f);
    const float beff = (conv_b[n] - bn_mean[n]) * sc + bn_beta[n];
    float weff[C_];
    #pragma unroll
    for (int c = 0; c < C_; ++c)
        weff[c] = 2.0f * conv_w[n * (C_ * 3) + c * 3 + 1] * sc;

    // B fragments (constant across T). B 4x16 layout: VGPR0 = K{0|2}, VGPR1 = K{1|3}.
    v2f bfrag1, bfrag2;
    bfrag1.x = half ? weff[2] : weff[0];
    bfrag1.y = half ? weff[3] : weff[1];
    bfrag2.x = half ? 0.0f    : weff[4];   // K=4 is channel 4, K=5..7 are zero pad
    bfrag2.y = 0.0f;

    const float fcw0 = fc_w[n];            // fc_w[0][n]
    const float fcw1 = fc_w[H_ + n];       // fc_w[1][n]
    const float fcb0 = fc_b[0];
    const float fcb1 = fc_b[1];
    const float b1s  = fminf(fmaxf(beta1p[0], 0.0f), 1.0f);
    const float b2s  = fminf(fmaxf(beta2p[0], 0.0f), 1.0f);

    v8f mem1 = {};
    float mem2a[8] = {}, mem2b[8] = {}, acca[8] = {}, accb[8] = {};

    // Async stage of one chunk: 320 rows of 64B, fully coalesced, direct to LDS.
    auto stage_async = [&](int t0, int buf) {
        #pragma unroll
        for (int k = 0; k < QUADS_PER_THREAD; ++k) {
            const int idx = tid + k * 128;     // 0..1279 float4 quads
            const int row = idx >> 2;          // 0..319 = bo*5 + c
            const int q   = idx & 3;           // float4 within 16-float row
            const int bo  = row / C_;
            const int c   = row - bo * C_;
            async_copy_b128(x + ((size_t)(bBlock + bo) * C_ + c) * T_ + t0 + q * 4,
                            &sx[buf][row * TSP + q * 4]);
        }
    };

    stage_async(0, 0);
    wait_async0();
    __syncthreads();

    for (int chunk = 0; chunk < NCHUNK; ++chunk) {
        const int buf = chunk & 1;
        if (chunk + 1 < NCHUNK)
            stage_async((chunk + 1) * TS, buf ^ 1);   // overlap next chunk's HBM traffic

        #pragma unroll 4
        for (int tt = 0; tt < TS; ++tt) {
            // A 16x4 f32 fragment: lane row M = n; half selects K pair {0,1} vs {2,3}.
            const float* xr = &sx[buf][(tb + n) * C_ * TSP + tt];
            v2f a1, a2;
            a1.x = xr[(half * 2 + 0) * TSP];
            a1.y = xr[(half * 2 + 1) * TSP];
            a2.x = half ? 0.0f : xr[4 * TSP];
            a2.y = 0.0f;

            v8f cacc;
            #pragma unroll
            for (int i = 0; i < 8; ++i) cacc[i] = beff;  // bias+BN folded into C init

            cacc = __builtin_amdgcn_wmma_f32_16x16x4_f32(
                       false, a1, false, bfrag1, (short)0, cacc, false, false);
            cacc = __builtin_amdgcn_wmma_f32_16x16x4_f32(
                       false, a2, false, bfrag2, (short)0, cacc, false, false);

            #pragma unroll
            for (int i = 0; i < 8; ++i) {
                // LIF layer 1 (element: batch m = i + 8*half, feature n)
                float m1  = mem1[i];
                float rst = (m1 > 1.0f) ? 1.0f : 0.0f;      // reset from OLD mem1
                m1 = b1s * m1 + cacc[i] - rst;
                mem1[i] = m1;
                const float spk = (m1 > 1.0f) ? 1.0f : 0.0f;

                // layer 2: cur2[m][o] = sum_n spk[m][n]*fc_w[o][n] + fc_b[o]
                // butterfly within each 16-lane half (row m is striped across it)
                float r0 = spk * fcw0;
                float r1 = spk * fcw1;
                r0 += __shfl_xor(r0, 1);  r1 += __shfl_xor(r1, 1);
                r0 += __shfl_xor(r0, 2);  r1 += __shfl_xor(r1, 2);
                r0 += __shfl_xor(r0, 4);  r1 += __shfl_xor(r1, 4);
                r0 += __shfl_xor(r0, 8);  r1 += __shfl_xor(r1, 8);
                const float cur0 = r0 + fcb0;
                const float cur1 = r1 + fcb1;

                // LIF layer 2 (state replicated across the 16-lane half)
                float m2a = mem2a[i];
                m2a = b2s * m2a + cur0 - ((m2a > 1.0f) ? 1.0f : 0.0f);
                mem2a[i] = m2a;
                acca[i] += (m2a > 1.0f) ? 1.0f : 0.0f;

                float m2b = mem2b[i];
                m2b = b2s * m2b + cur1 - ((m2b > 1.0f) ? 1.0f : 0.0f);
                mem2b[i] = m2b;
                accb[i] += (m2b > 1.0f) ? 1.0f : 0.0f;
            }
        }

        wait_async0();       // next chunk's LDS writes complete (per wave)
        __syncthreads();     // ... and visible to all waves before anyone reads them
    }

    // acc is replicated across each half; lanes 0 and 16 write their 8 rows.
    if (n == 0) {
        #pragma unroll
        for (int i = 0; i < 8; ++i) {
            const int m = i + half * 8;
            out[(size_t)(b0 + m) * O_ + 0] = acca[i];
            out[(size_t)(b0 + m) * O_ + 1] = accb[i];
        }
    }
}

extern "C" void kernel_launch(void* const* d_in, const int* in_sizes, int n_in,
                              void* d_out, int out_size, void* d_ws, size_t ws_size,
                              hipStream_t stream) {
    (void)in_sizes; (void)n_in; (void)out_size; (void)d_ws; (void)ws_size;
    snn_spike_kernel<<<B_ / 64, 128, 0, stream>>>(
        (const float*)d_in[0],   // x
        (const float*)d_in[1],   // conv_w
        (const float*)d_in[2],   // conv_b
        (const float*)d_in[3],   // bn_gamma
        (const float*)d_in[4],   // bn_beta
        (const float*)d_in[5],   // bn_mean
        (const float*)d_in[6],   // bn_var
        (const float*)d_in[7],   // fc_w
        (const float*)d_in[8],   // fc_b
        (const float*)d_in[9],   // beta1
        (const float*)d_in[10],  // beta2
        (float*)d_out);
}